// KDNet_Batch2_472446403143
// MI455X (gfx1250) — compile-verified
//
#include <hip/hip_runtime.h>
#include <hip/hip_bf16.h>

#define B_  32
#define CHN 8

typedef __attribute__((ext_vector_type(2))) float v2f;
typedef __attribute__((ext_vector_type(8))) float v8f;

// ---------------------------------------------------------------------------
// Level-0 input transpose: x [B, 8, 2048] -> xp [8, B*2048]
// ---------------------------------------------------------------------------
__global__ void kdnet_transpose_in(const float* __restrict__ x, float* __restrict__ xp) {
    int t = blockIdx.x * blockDim.x + threadIdx.x;
    const int total = B_ * CHN * 2048;
    if (t >= total) return;
    int l = t & 2047;
    int r = t >> 11;
    int c = r & 7;
    int b = r >> 3;
    xp[(size_t)c * (B_ * 2048) + b * 2048 + l] = x[t];
}

// ---------------------------------------------------------------------------
// Y[O x N] = W[O x K] * X[K x N], fp32 WMMA 16x16x4.
// block = 128 threads (4 waves); each wave computes a 64x32 tile
// (4 M-subtiles x 2 N-subtiles = 8 accumulators, 8 WMMAs per k-step).
// grid = (N/32, O/256). O % 256 == 0, N % 32 == 0, K % 4 == 0 guaranteed.
//
// A (16x4 f32, 2 VGPRs): lane&15 = M, (lane>>4)*2 = K base, comps K, K+1
// B (4x16 f32, 2 VGPRs): lane&15 = N, (lane>>4)*2 = K base, comps K, K+1
// C/D (16x16 f32, 8 VGPRs): comp v -> M = v + (lane>>4)*8, N = lane&15
// ---------------------------------------------------------------------------
__global__ void kdnet_gemm_wmma_f32(const float* __restrict__ W,
                                    const float* __restrict__ X,
                                    float* __restrict__ Y,
                                    int K, int N) {
    const int lane = threadIdx.x & 31;
    const int wave = threadIdx.x >> 5;
    const int n0   = blockIdx.x * 32;
    const int m0   = blockIdx.y * 256 + wave * 64;
    const int mrow = lane & 15;
    const int kb   = (lane >> 4) << 1;
    const int ncol = lane & 15;

    v8f acc00 = {}, acc10 = {}, acc20 = {}, acc30 = {};
    v8f acc01 = {}, acc11 = {}, acc21 = {}, acc31 = {};

    const float* Wp  = W + (size_t)(m0 + mrow) * K + kb;
    const float* Xp0 = X + (size_t)kb * N + n0 + ncol;
    const float* Xp1 = Xp0 + 16;

    for (int k0 = 0; k0 < K; k0 += 4) {
        v2f b0, b1;
        b0.x = Xp0[0];
        b0.y = Xp0[N];
        b1.x = Xp1[0];
        b1.y = Xp1[N];

        v2f a0, a1, a2, a3;
        a0.x = Wp[0];              a0.y = Wp[1];
        a1.x = Wp[(size_t)16 * K]; a1.y = Wp[(size_t)16 * K + 1];
        a2.x = Wp[(size_t)32 * K]; a2.y = Wp[(size_t)32 * K + 1];
        a3.x = Wp[(size_t)48 * K]; a3.y = Wp[(size_t)48 * K + 1];

        acc00 = __builtin_amdgcn_wmma_f32_16x16x4_f32(false, a0, false, b0, (short)0, acc00, false, false);
        acc10 = __builtin_amdgcn_wmma_f32_16x16x4_f32(false, a1, false, b0, (short)0, acc10, false, false);
        acc20 = __builtin_amdgcn_wmma_f32_16x16x4_f32(false, a2, false, b0, (short)0, acc20, false, false);
        acc30 = __builtin_amdgcn_wmma_f32_16x16x4_f32(false, a3, false, b0, (short)0, acc30, false, false);
        acc01 = __builtin_amdgcn_wmma_f32_16x16x4_f32(false, a0, false, b1, (short)0, acc01, false, false);
        acc11 = __builtin_amdgcn_wmma_f32_16x16x4_f32(false, a1, false, b1, (short)0, acc11, false, false);
        acc21 = __builtin_amdgcn_wmma_f32_16x16x4_f32(false, a2, false, b1, (short)0, acc21, false, false);
        acc31 = __builtin_amdgcn_wmma_f32_16x16x4_f32(false, a3, false, b1, (short)0, acc31, false, false);

        Wp  += 4;
        Xp0 += (size_t)4 * N;
        Xp1 += (size_t)4 * N;
    }

    const int mo = (lane >> 4) * 8;
    float* Yp0 = Y + (size_t)(m0 + mo) * N + n0 + ncol;
    float* Yp1 = Yp0 + 16;
#pragma unroll
    for (int v = 0; v < 8; v++) {
        Yp0[(size_t)(v +  0) * N] = acc00[v];
        Yp0[(size_t)(v + 16) * N] = acc10[v];
        Yp0[(size_t)(v + 32) * N] = acc20[v];
        Yp0[(size_t)(v + 48) * N] = acc30[v];
        Yp1[(size_t)(v +  0) * N] = acc01[v];
        Yp1[(size_t)(v + 16) * N] = acc11[v];
        Yp1[(size_t)(v + 32) * N] = acc21[v];
        Yp1[(size_t)(v + 48) * N] = acc31[v];
    }
}

// ---------------------------------------------------------------------------
// Per-output-channel batch stats: mean[o], istd[o] = rsqrt(var + eps)
// one block (256 threads) per channel; N = B*dim elements per row.
// ---------------------------------------------------------------------------
__global__ void kdnet_bn_stats(const float* __restrict__ Y,
                               float* __restrict__ mean,
                               float* __restrict__ istd,
                               int N, float inv_n) {
    const int o = blockIdx.x;
    const float* row = Y + (size_t)o * N;
    float s = 0.f, ss = 0.f;
    for (int i = threadIdx.x; i < N; i += blockDim.x) {
        float v = row[i];
        s += v;
        ss += v * v;
    }
    __shared__ float sh0[256];
    __shared__ float sh1[256];
    sh0[threadIdx.x] = s;
    sh1[threadIdx.x] = ss;
    __syncthreads();
    for (int st = 128; st > 0; st >>= 1) {
        if ((int)threadIdx.x < st) {
            sh0[threadIdx.x] += sh0[threadIdx.x + st];
            sh1[threadIdx.x] += sh1[threadIdx.x + st];
        }
        __syncthreads();
    }
    if (threadIdx.x == 0) {
        float mu  = sh0[0] * inv_n;
        float var = sh1[0] * inv_n - mu * mu;
        mean[o] = mu;
        istd[o] = rsqrtf(var + 1e-5f);
    }
}

// ---------------------------------------------------------------------------
// BN + LeakyReLU + cutdim gather + pair -> next x in [C2, B, dim/2] layout.
// next[c2, b, j2] = act( y_flat[b][ f*8*dim + j*8 + cutdim[b,j] ] )
//   with f = c2 mod featdim, j = 2*j2 + (c2 >= featdim)
// Y is stored [O, B*dim] with column n = b*dim + l.
// ---------------------------------------------------------------------------
__global__ void kdnet_bn_gather(const float* __restrict__ Y,
                                const float* __restrict__ mean,
                                const float* __restrict__ istd,
                                const float* __restrict__ gamma,
                                const float* __restrict__ beta,
                                const int*   __restrict__ cutdim,
                                float* __restrict__ Xn,
                                int featdim, int dim, int logdim) {
    int t = blockIdx.x * blockDim.x + threadIdx.x;
    const int dim2  = dim >> 1;
    const int total = featdim * B_ * dim;   // == 2*featdim * B * dim2
    if (t >= total) return;
    int j2 = t % dim2;
    int r  = t / dim2;
    int b  = r % B_;
    int c2 = r / B_;
    int hi = (c2 >= featdim) ? 1 : 0;
    int f  = c2 - hi * featdim;
    int j  = (j2 << 1) | hi;
    int cd = cutdim[b * dim + j];
    int F  = f * (CHN * dim) + j * CHN + cd;   // flat per-batch index into y[b]
    int o  = F >> logdim;
    int l  = F & (dim - 1);
    int N  = B_ * dim;
    float y = Y[(size_t)o * N + b * dim + l];
    float val = (y - mean[o]) * (istd[o] * gamma[o]) + beta[o];
    val = (val >= 0.f) ? val : 0.01f * val;
    Xn[t] = val;
}

// ---------------------------------------------------------------------------
// FC: out[b,k] = sum_c fw[k,c] * Xf[c*32 + b] + fb[k]   (Xf = [4096, 32])
// ---------------------------------------------------------------------------
__global__ void kdnet_fc(const float* __restrict__ Xf,
                         const float* __restrict__ fw,
                         const float* __restrict__ fb,
                         float* __restrict__ out) {
    int b = blockIdx.x;
    int k = threadIdx.x;
    if (k >= 40) return;
    float acc = 0.f;
    for (int c = 0; c < 4096; c++)
        acc = fmaf(fw[k * 4096 + c], Xf[c * B_ + b], acc);
    out[b * 40 + k] = acc + fb[k];
}

// ---------------------------------------------------------------------------
// Host launcher.
// d_in: [0]=x, [1..11]=conv_ws, [12..22]=bn_gammas, [23..33]=bn_betas,
//       [34]=fc_w, [35]=fc_b, [36..46]=cutdims (cutdims[j] has dim 2^(1+j)).
// Level i uses cutdims[10-i] = d_in[46 - i].
// Workspace: xA(16MB) | xB(16MB) | Y(128MB) | mean(64KB) | istd(64KB)
// ---------------------------------------------------------------------------
extern "C" void kernel_launch(void* const* d_in, const int* in_sizes, int n_in,
                              void* d_out, int out_size, void* d_ws, size_t ws_size,
                              hipStream_t stream) {
    (void)in_sizes; (void)n_in; (void)out_size; (void)ws_size;

    char* ws = (char*)d_ws;
    float* xA   = (float*)(ws);
    float* xB   = (float*)(ws + (size_t)16 * 1024 * 1024);
    float* Y    = (float*)(ws + (size_t)32 * 1024 * 1024);
    float* mean = (float*)(ws + (size_t)160 * 1024 * 1024);
    float* istd = (float*)(ws + (size_t)160 * 1024 * 1024 + 65536);

    static const int CH[11] = {64, 128, 128, 256, 256, 512, 512, 1024, 1024, 2048, 2048};

    // level-0 input into [C, B*L] layout
    kdnet_transpose_in<<<(B_ * CHN * 2048 + 255) / 256, 256, 0, stream>>>(
        (const float*)d_in[0], xA);

    float* cur = xA;
    float* nxt = xB;
    int inch = CHN;  // 8

    for (int i = 0; i < 11; i++) {
        const int dim    = 1 << (11 - i);
        const int logdim = 11 - i;
        const int featd  = CH[i];
        const int outch  = featd * CHN;
        const int N      = B_ * dim;

        const float* w   = (const float*)d_in[1 + i];
        const float* g   = (const float*)d_in[12 + i];
        const float* be  = (const float*)d_in[23 + i];
        const int*   cut = (const int*)d_in[46 - i];

        dim3 ggrid(N / 32, outch / 256);
        kdnet_gemm_wmma_f32<<<ggrid, 128, 0, stream>>>(w, cur, Y, inch, N);

        kdnet_bn_stats<<<outch, 256, 0, stream>>>(Y, mean, istd, N, 1.0f / (float)N);

        const int total = featd * B_ * dim;
        kdnet_bn_gather<<<(total + 255) / 256, 256, 0, stream>>>(
            Y, mean, istd, g, be, cut, nxt, featd, dim, logdim);

        float* tmp = cur; cur = nxt; nxt = tmp;
        inch = 2 * featd;
    }

    // after level 10: cur = [4096, B] ; FC -> [B, 40]
    kdnet_fc<<<B_, 64, 0, stream>>>(cur, (const float*)d_in[34],
                                    (const float*)d_in[35], (float*)d_out);
}